// SimpleGATModel_87943750353509
// MI455X (gfx1250) — compile-verified
//
#include <hip/hip_runtime.h>
#include <hip/hip_bf16.h>
#include <math.h>

#define F_IN 128
#define HIDC 16
#define HEADS 8
#define OUTC 40
#define NEG_SLOPE 0.2f

typedef float v2f __attribute__((ext_vector_type(2)));
typedef float v8f __attribute__((ext_vector_type(8)));

// ---- float <-> order-preserving unsigned (for atomic max on floats) ----
__device__ __forceinline__ unsigned f2ord(float f) {
  unsigned u = __float_as_uint(f);
  return (u & 0x80000000u) ? ~u : (u | 0x80000000u);
}
__device__ __forceinline__ float ord2f(unsigned u) {
  return (u & 0x80000000u) ? __uint_as_float(u & 0x7fffffffu) : __uint_as_float(~u);
}
__device__ __forceinline__ float leaky(float e) { return e > 0.f ? e : NEG_SLOPE * e; }

// ---------------- zero init ----------------
__global__ void gat_zero_kernel(float* __restrict__ p, size_t n) {
  size_t i = (size_t)blockIdx.x * blockDim.x + threadIdx.x;
  if (i < n) p[i] = 0.f;
}

// ---------------- f32 WMMA GEMM: C[M,Nc] = A[M,K] @ B[K,Nc], row-major ----
// One wave per 16x16 output tile; V_WMMA_F32_16X16X4_F32, K stepped by 4.
// Fragment layout (wave32): lane = 16*g + l16.
//   A (16x4):  a[v] = A[l16][2g+v]       (two consecutive K per lane)
//   B (4x16):  b[v] = B[2g+v][l16]
//   C (16x16): acc[v] = C[8g+v][l16]
__global__ void gat_gemm_wmma_f32(const float* __restrict__ A, const float* __restrict__ B,
                                  float* __restrict__ C, int M, int K, int Nc) {
  int tile = blockIdx.x * blockDim.y + threadIdx.y;   // wave-uniform
  int ntn = Nc >> 4;
  int tm = tile / ntn, tn = tile - tm * ntn;
  if (tm * 16 >= M) return;                           // wave-uniform branch
  int lane = threadIdx.x, g = lane >> 4, l16 = lane & 15;
  v8f acc = {};
  const float* arow = A + (size_t)(tm * 16 + l16) * K + 2 * g;
  const float* bcol = B + (size_t)(2 * g) * Nc + tn * 16 + l16;
  for (int k = 0; k < K; k += 4) {
    v2f a, b;
    a.x = arow[k];
    a.y = arow[k + 1];
    b.x = bcol[(size_t)k * Nc];
    b.y = bcol[(size_t)(k + 1) * Nc];
    acc = __builtin_amdgcn_wmma_f32_16x16x4_f32(false, a, false, b, (short)0, acc,
                                                false, false);
  }
  float* crow = C + (size_t)(tm * 16 + 8 * g) * Nc + tn * 16 + l16;
#pragma unroll
  for (int v = 0; v < 8; ++v) crow[(size_t)v * Nc] = acc[v];
}

// ---------------- per-(node,head) attention projections ----------------
template <int H, int C>
__global__ void gat_alpha_kernel(const float* __restrict__ h, const float* __restrict__ att_s,
                                 const float* __restrict__ att_d, float* __restrict__ as_,
                                 float* __restrict__ ad_, int Nn) {
  int i = blockIdx.x * blockDim.x + threadIdx.x;  // i = n*H + hh
  if (i >= Nn * H) return;
  int hh = i % H;
  const float* hp = h + (size_t)i * C;            // n*H*C + hh*C == i*C
  float s = 0.f, d = 0.f;
#pragma unroll
  for (int c = 0; c < C; ++c) {
    float v = hp[c];
    s += v * att_s[hh * C + c];
    d += v * att_d[hh * C + c];
  }
  as_[i] = s;
  ad_[i] = d;
}

// ---------------- segment softmax pass 1: per-dst max ----------------
template <int H>
__global__ void gat_edge_max_kernel(const int* __restrict__ ei, int E, int Nn,
                                    const float* __restrict__ as_, const float* __restrict__ ad_,
                                    unsigned* __restrict__ mord) {
  int i = blockIdx.x * blockDim.x + threadIdx.x;
  if (i >= E + Nn) return;
  int src = (i < E) ? ei[i] : (i - E);        // self-loops appended
  int dst = (i < E) ? ei[E + i] : (i - E);
#pragma unroll
  for (int hh = 0; hh < H; ++hh) {
    float e = leaky(as_[src * H + hh] + ad_[dst * H + hh]);
    atomicMax(&mord[dst * H + hh], f2ord(e));
  }
}

// ---------------- segment softmax pass 2: per-dst sum of exp ----------------
template <int H>
__global__ void gat_edge_sum_kernel(const int* __restrict__ ei, int E, int Nn,
                                    const float* __restrict__ as_, const float* __restrict__ ad_,
                                    const unsigned* __restrict__ mord, float* __restrict__ ssum) {
  int i = blockIdx.x * blockDim.x + threadIdx.x;
  if (i >= E + Nn) return;
  int src = (i < E) ? ei[i] : (i - E);
  int dst = (i < E) ? ei[E + i] : (i - E);
#pragma unroll
  for (int hh = 0; hh < H; ++hh) {
    float e = leaky(as_[src * H + hh] + ad_[dst * H + hh]);
    atomicAdd(&ssum[dst * H + hh], __expf(e - ord2f(mord[dst * H + hh])));
  }
}

// ---------------- message pass: one thread per (edge, head) ----------------
template <int H, int C>
__global__ void gat_edge_msg_kernel(const int* __restrict__ ei, int E, int Nn,
                                    const float* __restrict__ as_, const float* __restrict__ ad_,
                                    const unsigned* __restrict__ mord, const float* __restrict__ ssum,
                                    const float* __restrict__ h, float* __restrict__ out) {
  int t = blockIdx.x * blockDim.x + threadIdx.x;
  if (t >= (E + Nn) * H) return;
  int i = t / H, hh = t - i * H;
  int src = (i < E) ? ei[i] : (i - E);
  int dst = (i < E) ? ei[E + i] : (i - E);
  float e = leaky(as_[src * H + hh] + ad_[dst * H + hh]);
  float alpha = __expf(e - ord2f(mord[dst * H + hh])) / (ssum[dst * H + hh] + 1e-16f);
  const float* hp = h + ((size_t)src * H + hh) * C;
  float* op = out + ((size_t)dst * H + hh) * C;
#pragma unroll
  for (int c = 0; c < C; ++c) atomicAdd(&op[c], hp[c] * alpha);
}

// ---------------- bias + ReLU (GAT1 epilogue) ----------------
template <int W>
__global__ void gat_bias_relu_kernel(float* __restrict__ x, const float* __restrict__ b, size_t n) {
  size_t i = (size_t)blockIdx.x * blockDim.x + threadIdx.x;
  if (i >= n) return;
  x[i] = fmaxf(x[i] + b[i % W], 0.f);
}

// ---------------- final: (+b2) @ Wl + bl, row softmax ----------------
__global__ void gat_final_kernel(const float* __restrict__ h2, const float* __restrict__ b2,
                                 const float* __restrict__ Wl, const float* __restrict__ bl,
                                 float* __restrict__ y, int Nn) {
  int n = blockIdx.x * blockDim.x + threadIdx.x;
  if (n >= Nn) return;
  float hr[HIDC];
#pragma unroll
  for (int c = 0; c < HIDC; ++c) hr[c] = h2[(size_t)n * HIDC + c] + b2[c];
  float lg[OUTC];
  float mx = -INFINITY;
  for (int o = 0; o < OUTC; ++o) {
    float v = bl[o];
#pragma unroll
    for (int c = 0; c < HIDC; ++c) v += hr[c] * Wl[c * OUTC + o];
    lg[o] = v;
    mx = fmaxf(mx, v);
  }
  float s = 0.f;
  for (int o = 0; o < OUTC; ++o) {
    lg[o] = __expf(lg[o] - mx);
    s += lg[o];
  }
  float inv = 1.f / s;
  for (int o = 0; o < OUTC; ++o) y[(size_t)n * OUTC + o] = lg[o] * inv;
}

extern "C" void kernel_launch(void* const* d_in, const int* in_sizes, int n_in,
                              void* d_out, int out_size, void* d_ws, size_t ws_size,
                              hipStream_t stream) {
  const float* x    = (const float*)d_in[0];
  const int*   ei   = (const int*)d_in[1];
  const float* W1   = (const float*)d_in[2];
  const float* ats1 = (const float*)d_in[3];
  const float* atd1 = (const float*)d_in[4];
  const float* b1   = (const float*)d_in[5];
  const float* W2   = (const float*)d_in[6];
  const float* ats2 = (const float*)d_in[7];
  const float* atd2 = (const float*)d_in[8];
  const float* b2   = (const float*)d_in[9];
  const float* Wl   = (const float*)d_in[10];
  const float* bl   = (const float*)d_in[11];

  const int Nn = in_sizes[0] / F_IN;   // 100000
  const int Ee = in_sizes[1] / 2;      // 1600000
  const int ET = Ee + Nn;              // with self-loops

  // ---- workspace carve (floats) ----
  float* p = (float*)d_ws;
  float* h1   = p; p += (size_t)Nn * F_IN;          // x @ W1            [N,128]
  float* as1  = p; p += (size_t)Nn * HEADS;         // alpha_src GAT1    [N,8]
  float* ad1  = p; p += (size_t)Nn * HEADS;         // alpha_dst GAT1    [N,8]
  float* h2   = p; p += (size_t)Nn * HIDC;          // relu(out1) @ W2   [N,16]
  float* as2  = p; p += (size_t)Nn;                 // alpha_src GAT2    [N]
  float* ad2  = p; p += (size_t)Nn;                 // alpha_dst GAT2    [N]
  float* zb   = p;                                  // ---- zero-init region ----
  unsigned* m1 = (unsigned*)p; p += (size_t)Nn * HEADS;  // ordered-uint max [N,8]
  float* s1   = p; p += (size_t)Nn * HEADS;         // exp-sum GAT1      [N,8]
  float* out1 = p; p += (size_t)Nn * F_IN;          // aggregated GAT1   [N,128]
  unsigned* m2 = (unsigned*)p; p += (size_t)Nn;     // ordered-uint max  [N]
  float* s2   = p; p += (size_t)Nn;                 // exp-sum GAT2      [N]
  float* out2 = p; p += (size_t)Nn * HIDC;          // aggregated GAT2   [N,16]
  const size_t zcount = (size_t)Nn * (HEADS + HEADS + F_IN + 1 + 1 + HIDC); // 162*N

  const int BS = 256;
  // 0) zero accumulators + ordered-max inits (0 < f2ord(-inf), so 0 is identity)
  gat_zero_kernel<<<dim3((unsigned)((zcount + BS - 1) / BS)), dim3(BS), 0, stream>>>(zb, zcount);

  const int tilesM = (Nn + 15) / 16;
  // 1) h1 = x @ W1   [N,128]x[128,128]
  {
    int tiles = tilesM * (F_IN / 16);
    gat_gemm_wmma_f32<<<dim3((tiles + 3) / 4), dim3(32, 4), 0, stream>>>(x, W1, h1, Nn, F_IN, F_IN);
  }
  // 2) attention projections for GAT1
  gat_alpha_kernel<HEADS, HIDC><<<dim3((Nn * HEADS + BS - 1) / BS), dim3(BS), 0, stream>>>(
      h1, ats1, atd1, as1, ad1, Nn);
  // 3) segment softmax + aggregation (GAT1)
  gat_edge_max_kernel<HEADS><<<dim3((ET + BS - 1) / BS), dim3(BS), 0, stream>>>(ei, Ee, Nn, as1, ad1, m1);
  gat_edge_sum_kernel<HEADS><<<dim3((ET + BS - 1) / BS), dim3(BS), 0, stream>>>(ei, Ee, Nn, as1, ad1, m1, s1);
  gat_edge_msg_kernel<HEADS, HIDC><<<dim3((ET * HEADS + BS - 1) / BS), dim3(BS), 0, stream>>>(
      ei, Ee, Nn, as1, ad1, m1, s1, h1, out1);
  // 4) out1 = relu(out1 + b1)
  gat_bias_relu_kernel<F_IN><<<dim3((unsigned)(((size_t)Nn * F_IN + BS - 1) / BS)), dim3(BS), 0, stream>>>(
      out1, b1, (size_t)Nn * F_IN);
  // 5) h2 = out1 @ W2   [N,128]x[128,16]
  {
    int tiles = tilesM * (HIDC / 16);
    gat_gemm_wmma_f32<<<dim3((tiles + 3) / 4), dim3(32, 4), 0, stream>>>(out1, W2, h2, Nn, F_IN, HIDC);
  }
  // 6) attention projections for GAT2 (1 head)
  gat_alpha_kernel<1, HIDC><<<dim3((Nn + BS - 1) / BS), dim3(BS), 0, stream>>>(h2, ats2, atd2, as2, ad2, Nn);
  // 7) segment softmax + aggregation (GAT2)
  gat_edge_max_kernel<1><<<dim3((ET + BS - 1) / BS), dim3(BS), 0, stream>>>(ei, Ee, Nn, as2, ad2, m2);
  gat_edge_sum_kernel<1><<<dim3((ET + BS - 1) / BS), dim3(BS), 0, stream>>>(ei, Ee, Nn, as2, ad2, m2, s2);
  gat_edge_msg_kernel<1, HIDC><<<dim3((ET + BS - 1) / BS), dim3(BS), 0, stream>>>(
      ei, Ee, Nn, as2, ad2, m2, s2, h2, out2);
  // 8) logits = (out2 + b2) @ Wl + bl, softmax -> d_out
  gat_final_kernel<<<dim3((Nn + BS - 1) / BS), dim3(BS), 0, stream>>>(out2, b2, Wl, bl, (float*)d_out, Nn);
}